// ExcitationShaper_30717606101139
// MI455X (gfx1250) — compile-verified
//
#include <hip/hip_runtime.h>
#include <stdint.h>

// ----------------------------------------------------------------------------
// ExcitationShaper for MI455X (gfx1250), wave32.
//
// Pipeline (all on `stream`, graph-capture safe):
//   memset  : zero segment-sum table (atomicAdd targets)
//   k1      : per-chunk onset counts
//   k2      : per-batch exclusive scan of chunk counts (seg-id bases)
//   k3      : per-chunk walk -> seg_ids + atomic segment sums/counts
//   k4      : elementwise params->(w,q) mapping + fractional comb  -> xc
//   k6      : chunked affine composition of the time-varying biquad (pass A)
//             inputs staged by ONE Tensor-Data-Mover 2D tile load (48 KB)
//   k7      : per-batch scan of chunk transforms -> chunk init states (pass B)
//   k8      : chunk replay -> y (pass C); TDM 2D load in, TDM 1D store out
// ----------------------------------------------------------------------------

#define BB   32
#define TT   65536
#define LL   128                 // samples per chunk
#define CPB  (TT / LL)           // 512 chunks per batch row
#define NCH  (BB * CPB)          // 16384 chunks total
#define NSEG (TT + 1)
#define REG  (32 * LL)           // 4096 floats staged per wave per array

#define LOG20  2.9957322735539909f   // log(2.0/0.1)
#define LOG400 5.9914645471079817f   // log(pi / min_w), min_w = 2*pi*20/16000
#define MINW   0.0078539816339744831f
#define MIND   0.1f
#define MINQ   0.1f

typedef __attribute__((ext_vector_type(4))) unsigned int v4u;
typedef __attribute__((ext_vector_type(4))) int          v4i;
typedef __attribute__((ext_vector_type(8))) int          v8i;

__device__ __forceinline__ float sigm(float v) { return 1.0f / (1.0f + __expf(-v)); }

// ---------------------------------------------------------------- k1: counts
__global__ void k1_chunk_onsets(const int* __restrict__ onsets, int* __restrict__ cntC) {
    int gid = blockIdx.x * blockDim.x + threadIdx.x;     // chunk id = b*CPB + c
    const int* p = onsets + (size_t)gid * LL;            // flat b*TT + c*LL == gid*LL
    int s = 0;
#pragma unroll 4
    for (int j = 0; j < LL; ++j) s += p[j];
    cntC[gid] = s;
}

// -------------------------------------------------- k2: per-batch excl. scan
__global__ void k2_scan_chunks(const int* __restrict__ cntC, int* __restrict__ baseC) {
    int b = threadIdx.x;                                 // 32 threads, one per batch
    int run = 0;
    for (int c = 0; c < CPB; ++c) {
        baseC[b * CPB + c] = run;
        run += cntC[b * CPB + c];
    }
}

// ------------------------------------- k3: seg ids + atomic segment sums/cnt
__global__ void k3_segids_sums(const int* __restrict__ onsets,
                               const float* __restrict__ params,
                               const int* __restrict__ baseC,
                               int* __restrict__ segid,
                               float* __restrict__ segsum) {
    int gid = blockIdx.x * blockDim.x + threadIdx.x;     // chunk id
    int b   = gid >> 9;                                  // CPB = 512
    int seg = baseC[gid];
    size_t t0 = (size_t)gid * LL;
    float* s0 = segsum + 0 * (size_t)BB * NSEG + (size_t)b * NSEG;
    float* s1 = segsum + 1 * (size_t)BB * NSEG + (size_t)b * NSEG;
    float* s2 = segsum + 2 * (size_t)BB * NSEG + (size_t)b * NSEG;
    float* s3 = segsum + 3 * (size_t)BB * NSEG + (size_t)b * NSEG;
    float* sc = segsum + 4 * (size_t)BB * NSEG + (size_t)b * NSEG;
    for (int j = 0; j < LL; ++j) {
        size_t t = t0 + j;
        seg += onsets[t];                                // inclusive cumsum
        segid[t] = seg;
        const float* pp = params + t * 4;
        atomicAdd(&s0[seg], pp[0]);
        atomicAdd(&s1[seg], pp[1]);
        atomicAdd(&s2[seg], pp[2]);
        atomicAdd(&s3[seg], pp[3]);
        atomicAdd(&sc[seg], 1.0f);
    }
}

// helper: distance at an arbitrary flat index (recomputed from segment means)
__device__ __forceinline__ float seg_dist(const int* __restrict__ segid,
                                          const float* __restrict__ s0,
                                          const float* __restrict__ sc, int flat) {
    int sg  = segid[flat];
    float c = fmaxf(sc[sg], 1.0f);
    return MIND * __expf(sigm(s0[sg] / c) * LOG20);
}

// -------------------------------- k4: param mapping + fractional-delay comb
__global__ void k4_params_comb(const float* __restrict__ f0,
                               const float* __restrict__ x,
                               const int* __restrict__ segid,
                               const float* __restrict__ segsum,
                               float* __restrict__ warr,
                               float* __restrict__ qarr,
                               float* __restrict__ xcarr) {
    int gid = blockIdx.x * blockDim.x + threadIdx.x;     // over B*T
    int b = gid >> 16;
    int t = gid & (TT - 1);
    const float* s0 = segsum + 0 * (size_t)BB * NSEG + (size_t)b * NSEG;
    const float* s1 = segsum + 1 * (size_t)BB * NSEG + (size_t)b * NSEG;
    const float* s2 = segsum + 2 * (size_t)BB * NSEG + (size_t)b * NSEG;
    const float* s3 = segsum + 3 * (size_t)BB * NSEG + (size_t)b * NSEG;
    const float* sc = segsum + 4 * (size_t)BB * NSEG + (size_t)b * NSEG;

    int sg = segid[gid];
    float inv = 1.0f / fmaxf(sc[sg], 1.0f);
    float m0 = s0[sg] * inv, m1 = s1[sg] * inv, m2 = s2[sg] * inv, m3 = s3[sg] * inv;

    float dist = MIND * __expf(sigm(m0) * LOG20);
    float wv   = MINW * __expf(sigm(m1) * LOG400);
    float qv   = MINQ * __expf(sigm(m2) * LOG20);
    float mu   = sigm(m3);
    warr[gid] = wv;
    qarr[gid] = qv;

    float p = f0[gid] * mu;                              // fractional pluck pos
    float z = floorf(p);
    float alfa = p - z;
    int zi = (int)z;
    int rowbase = b << 16;

    float xdt = x[gid] * dist;
    int i1 = t - (zi + 1), i2 = t - (zi + 2);
    float g1 = 0.0f, g2 = 0.0f;
    if (i1 >= 0) g1 = x[rowbase + i1] * seg_dist(segid, s0, sc, rowbase + i1);
    if (i2 >= 0) g2 = x[rowbase + i2] * seg_dist(segid, s0, sc, rowbase + i2);
    xcarr[gid] = xdt - (1.0f - alfa) * g1 - alfa * g2;
}

// ----------------------------------------------------------------------------
// Tensor Data Mover descriptors (CDNA5 ISA).
// Toolchain arity differs: clang-22 (ROCm 7.2) = 5 args, clang-23 = 6 args.
// ----------------------------------------------------------------------------
#if defined(__clang_major__) && (__clang_major__ >= 23)
#define TDM_LOAD(g0, g1, g2, g3)  __builtin_amdgcn_tensor_load_to_lds((g0), (g1), (g2), (g3), (v8i)(0), 0)
#define TDM_STORE(g0, g1, g2, g3) __builtin_amdgcn_tensor_store_from_lds((g0), (g1), (g2), (g3), (v8i)(0), 0)
#else
#define TDM_LOAD(g0, g1, g2, g3)  __builtin_amdgcn_tensor_load_to_lds((g0), (g1), (g2), (g3), 0)
#define TDM_STORE(g0, g1, g2, g3) __builtin_amdgcn_tensor_store_from_lds((g0), (g1), (g2), (g3), 0)
#endif

// group0: count=1 | lds_addr | global_addr[56:0] | type=2 ("image")
__device__ __forceinline__ v4u tdm_g0(unsigned ldsAddr, const void* g) {
    unsigned long long ga = (unsigned long long)(uintptr_t)g;
    v4u g0;
    g0.x = 1u;
    g0.y = ldsAddr;
    g0.z = (unsigned)(ga & 0xFFFFFFFFu);
    g0.w = (unsigned)((ga >> 32) & 0x1FFFFFFu) | (2u << 30);
    return g0;
}

// group1: data_size=2 (4B), 2D tile {tile0, tile1}, tensor {td0, td1}, stride0
__device__ __forceinline__ v8i tdm_g1(unsigned td0, unsigned td1,
                                      unsigned tile0, unsigned tile1,
                                      unsigned stride0) {
    v8i g1;
    g1[0] = (int)(2u << 16);                             // data_size = 4 bytes
    g1[1] = (int)((td0 & 0xFFFFu) << 16);                // tensor_dim0[15:0]
    g1[2] = (int)(((td0 >> 16) & 0xFFFFu) | ((td1 & 0xFFFFu) << 16));
    g1[3] = (int)(((td1 >> 16) & 0xFFFFu) | ((tile0 & 0xFFFFu) << 16));
    g1[4] = (int)(tile1 & 0xFFFFu);                      // tile_dim1, tile_dim2=0
    g1[5] = (int)stride0;                                // tensor_dim0_stride[31:0]
    g1[6] = 0;                                           // stride0 hi | stride1 lo
    g1[7] = 0;
    return g1;
}

// One TDM load stages 3 rows x 4096 floats (w | q | xc), row stride BB*TT.
__device__ __forceinline__ void tdm_load_wqx(unsigned ldsAddr, const float* gbase) {
    v4u g0 = tdm_g0(ldsAddr, gbase);
    v8i g1 = tdm_g1(/*td0*/REG, /*td1*/3, /*tile0*/REG, /*tile1*/3,
                    /*stride0*/(unsigned)(BB * TT));
    v4i gz = {0, 0, 0, 0};
    TDM_LOAD(g0, g1, gz, gz);
}

// One TDM store flushes 4096 contiguous floats from LDS to global.
__device__ __forceinline__ void tdm_store_row(unsigned ldsAddr, float* gbase) {
    v4u g0 = tdm_g0(ldsAddr, gbase);
    v8i g1 = tdm_g1(/*td0*/REG, /*td1*/1, /*tile0*/REG, /*tile1*/1, /*stride0*/REG);
    v4i gz = {0, 0, 0, 0};
    TDM_STORE(g0, g1, gz, gz);
}

__device__ __forceinline__ void biquad_coefs(float wv, float qv,
                                             float& b0, float& b1c, float& a1, float& a2) {
    float swv, cwv;
    __sincosf(wv, &swv, &cwv);
    float alpha = swv / (2.0f * qv);
    float inv = 1.0f / (1.0f + alpha);
    b0  = (1.0f - cwv) * 0.5f * inv;                     // b2 == b0
    b1c = (1.0f - cwv) * inv;
    a1  = -2.0f * cwv * inv;
    a2  = (1.0f - alpha) * inv;
}

// -------------------- k6: pass A — compose chunk affine transform (M, d)
// wqx = base of contiguous [warr | qarr | xcarr] planes, each BB*TT floats.
__global__ __launch_bounds__(32) void k6_chunk_affine(const float* __restrict__ wqx,
                                                      float* __restrict__ chunkM) {
    __shared__ __align__(16) float smem[3 * REG];        // 48 KB: w | q | xc
    int lane = threadIdx.x;
    int gid  = blockIdx.x * 32 + lane;                   // chunk id
    size_t regionBase = (size_t)blockIdx.x * REG;
    unsigned lb = (unsigned)(uintptr_t)smem;

    tdm_load_wqx(lb, wqx + regionBase);                  // 48 KB in one DMA
    asm volatile("s_wait_tensorcnt 0x0" ::: "memory");

    const float* sw = smem;
    const float* sq = smem + REG;
    const float* sx = smem + 2 * REG;
    int lbase = lane * LL;

    float m00 = 1.f, m01 = 0.f, m10 = 0.f, m11 = 1.f, d0 = 0.f, d1 = 0.f;
    for (int j = 0; j < LL; ++j) {
        float b0, b1c, a1, a2;
        biquad_coefs(sw[lbase + j], sq[lbase + j], b0, b1c, a1, a2);
        float xv = sx[lbase + j];
        float c0 = (b1c - a1 * b0) * xv;
        float c1 = (b0  - a2 * b0) * xv;
        float na1 = -a1, na2 = -a2;
        float nm00 = na1 * m00 + m10;
        float nm01 = na1 * m01 + m11;
        float nm10 = na2 * m00;
        float nm11 = na2 * m01;
        float nd0  = na1 * d0 + d1 + c0;
        float nd1  = na2 * d0 + c1;
        m00 = nm00; m01 = nm01; m10 = nm10; m11 = nm11; d0 = nd0; d1 = nd1;
    }
    float* o = chunkM + (size_t)gid * 6;
    o[0] = m00; o[1] = m01; o[2] = m10; o[3] = m11; o[4] = d0; o[5] = d1;
}

// ------------- k7: pass B — per-batch scan of chunk transforms -> init states
__global__ void k7_scan_states(const float* __restrict__ chunkM, float* __restrict__ initS) {
    int b = threadIdx.x;                                 // 32 threads
    float s1 = 0.f, s2 = 0.f;
    for (int c = 0; c < CPB; ++c) {
        int gid = b * CPB + c;
        initS[gid * 2]     = s1;
        initS[gid * 2 + 1] = s2;
        const float* m = chunkM + (size_t)gid * 6;
        float n1 = m[0] * s1 + m[1] * s2 + m[4];
        float n2 = m[2] * s1 + m[3] * s2 + m[5];
        s1 = n1; s2 = n2;
    }
}

// ---------------- k8: pass C — replay chunk from init state, emit y via TDM
__global__ __launch_bounds__(32) void k8_biquad_out(const float* __restrict__ wqx,
                                                    const float* __restrict__ initS,
                                                    float* __restrict__ out) {
    __shared__ __align__(16) float smem[3 * REG];
    int lane = threadIdx.x;
    int gid  = blockIdx.x * 32 + lane;
    size_t regionBase = (size_t)blockIdx.x * REG;
    unsigned lb = (unsigned)(uintptr_t)smem;

    tdm_load_wqx(lb, wqx + regionBase);
    asm volatile("s_wait_tensorcnt 0x0" ::: "memory");

    const float* sw = smem;
    const float* sq = smem + REG;
    float*       sx = smem + 2 * REG;                    // reused for outputs
    int lbase = lane * LL;

    float s1 = initS[gid * 2], s2 = initS[gid * 2 + 1];
    for (int j = 0; j < LL; ++j) {
        float b0, b1c, a1, a2;
        biquad_coefs(sw[lbase + j], sq[lbase + j], b0, b1c, a1, a2);
        float xv = sx[lbase + j];
        float y   = b0 * xv + s1;
        float ns1 = b1c * xv - a1 * y + s2;
        float ns2 = b0  * xv - a2 * y;                   // b2 == b0
        s1 = ns1; s2 = ns2;
        sx[lbase + j] = y;                               // stage output in place
    }

    // LDS writes must land before the TDM engine reads them, then DMA out.
    asm volatile("s_wait_dscnt 0x0" ::: "memory");
    tdm_store_row(lb + 2u * REG * 4u, out + regionBase);
    asm volatile("s_wait_tensorcnt 0x0" ::: "memory");
}

// ----------------------------------------------------------------------------
extern "C" void kernel_launch(void* const* d_in, const int* in_sizes, int n_in,
                              void* d_out, int out_size, void* d_ws, size_t ws_size,
                              hipStream_t stream) {
    (void)in_sizes; (void)n_in; (void)out_size; (void)ws_size;
    const float* f0     = (const float*)d_in[0];
    const float* x      = (const float*)d_in[1];
    const float* params = (const float*)d_in[2];
    const int*   onsets = (const int*)d_in[3];
    float* out = (float*)d_out;

    // Workspace carve-out (~76 MB, every offset 16 B aligned).
    // NOTE: warr/qarr/xcarr MUST stay contiguous in this order — the TDM
    // 2D tile descriptor uses row stride BB*TT floats across the 3 planes.
    char* w = (char*)d_ws;
    float* segsum = (float*)w;  w += (size_t)5 * BB * NSEG * sizeof(float);
    int*   segid  = (int*)w;    w += (size_t)BB * TT * sizeof(int);
    int*   cntC   = (int*)w;    w += (size_t)NCH * sizeof(int);
    int*   baseC  = (int*)w;    w += (size_t)NCH * sizeof(int);
    float* warr   = (float*)w;  w += (size_t)BB * TT * sizeof(float);
    float* qarr   = (float*)w;  w += (size_t)BB * TT * sizeof(float);
    float* xcarr  = (float*)w;  w += (size_t)BB * TT * sizeof(float);
    float* chunkM = (float*)w;  w += (size_t)NCH * 6 * sizeof(float);
    float* initS  = (float*)w;

    hipMemsetAsync(segsum, 0, (size_t)5 * BB * NSEG * sizeof(float), stream);
    k1_chunk_onsets<<<NCH / 256, 256, 0, stream>>>(onsets, cntC);
    k2_scan_chunks<<<1, BB, 0, stream>>>(cntC, baseC);
    k3_segids_sums<<<NCH / 256, 256, 0, stream>>>(onsets, params, baseC, segid, segsum);
    k4_params_comb<<<(BB * TT) / 256, 256, 0, stream>>>(f0, x, segid, segsum,
                                                        warr, qarr, xcarr);
    k6_chunk_affine<<<NCH / 32, 32, 0, stream>>>(warr, chunkM);
    k7_scan_states<<<1, BB, 0, stream>>>(chunkM, initS);
    k8_biquad_out<<<NCH / 32, 32, 0, stream>>>(warr, initS, out);
}